// SEAL_77498389889831
// MI455X (gfx1250) — compile-verified
//
#include <hip/hip_runtime.h>

// ---------------------------------------------------------------------------
// GCN 3-layer forward (eval) for gfx1250 / MI455X.
//   deg -> dinv  (f32 atomics + v_rsq)
//   per layer: WMMA f32 GEMM (16x16x4), zero agg, edge scatter-add
//              (global_atomic_add_f32), fused self-loop + bias + BN + ReLU.
// Working set (~77MB) is L2-resident (192MB); scatter is the bottleneck so
// everything stays fp32 and atomics are the hardware fadd path.
// ---------------------------------------------------------------------------

typedef __attribute__((ext_vector_type(2))) float v2f;
typedef __attribute__((ext_vector_type(8))) float v8f;

#define BN_EPS 1e-5f

// ---- zero a buffer (count is in float4 quads; all our sizes are /4) -------
__global__ void zero_f32_kernel(float* __restrict__ p, long long n4) {
  long long t = (long long)blockIdx.x * blockDim.x + threadIdx.x;
  if (t < n4) reinterpret_cast<float4*>(p)[t] = make_float4(0.f, 0.f, 0.f, 0.f);
}

// ---- degree: deg[dst] += 1 per edge (float to match segment_sum of ones) --
__global__ void deg_kernel(const long long* __restrict__ ei,
                           float* __restrict__ deg, long long E) {
  long long e = (long long)blockIdx.x * blockDim.x + threadIdx.x;
  if (e < E) unsafeAtomicAdd(&deg[ei[E + e]], 1.0f);
}

// ---- dinv = rsqrt(deg + 1) in place ---------------------------------------
__global__ void dinv_kernel(float* __restrict__ d, int n) {
  int i = blockIdx.x * blockDim.x + threadIdx.x;
  if (i < n) d[i] = rsqrtf(d[i] + 1.0f);
}

// ---- WMMA f32 GEMM: H[N][64] = X[N][FIN] @ W[FIN][64] ---------------------
// One wave computes a 16-row x 64-col stripe (4 x v8f accumulators).
// A frag  (16x4 f32): lane L -> M = L%16, K pair = 2*(L/16)   [ISA 7.12.2]
// B frag  (4x16 f32): lane L -> N = L%16, K pair = 2*(L/16)   (mirror of A)
// D (16x16 f32):      VGPR i -> row i + 8*(L/16), col L%16
template <int FIN>
__global__ void gcn_gemm_kernel(const float* __restrict__ X,
                                const float* __restrict__ W,
                                float* __restrict__ H, int nrows) {
  const int wave = blockIdx.x * (blockDim.x >> 5) + (threadIdx.x >> 5);
  const int lane = threadIdx.x & 31;
  const int row0 = wave * 16;
  if (row0 + 16 > nrows) return;              // wave-uniform: EXEC stays all-1s

  const int m    = lane & 15;                 // row (A) / col (B) within tile
  const int half = lane >> 4;                 // 0 or 1
  const int kq   = half * 2;                  // K offset inside 4-step

  const float* __restrict__ xrow = X + (size_t)(row0 + m) * FIN;
  v8f acc0 = {}, acc1 = {}, acc2 = {}, acc3 = {};

#pragma unroll 4
  for (int k0 = 0; k0 < FIN; k0 += 4) {
    v2f a;
    a.x = xrow[k0 + kq];
    a.y = xrow[k0 + kq + 1];
    const float* __restrict__ wk0 = W + (size_t)(k0 + kq) * 64;
    const float* __restrict__ wk1 = wk0 + 64;
    v2f b0, b1, b2, b3;
    b0.x = wk0[m];      b0.y = wk1[m];
    b1.x = wk0[16 + m]; b1.y = wk1[16 + m];
    b2.x = wk0[32 + m]; b2.y = wk1[32 + m];
    b3.x = wk0[48 + m]; b3.y = wk1[48 + m];
    acc0 = __builtin_amdgcn_wmma_f32_16x16x4_f32(false, a, false, b0, (short)0, acc0, false, false);
    acc1 = __builtin_amdgcn_wmma_f32_16x16x4_f32(false, a, false, b1, (short)0, acc1, false, false);
    acc2 = __builtin_amdgcn_wmma_f32_16x16x4_f32(false, a, false, b2, (short)0, acc2, false, false);
    acc3 = __builtin_amdgcn_wmma_f32_16x16x4_f32(false, a, false, b3, (short)0, acc3, false, false);
  }

  float* __restrict__ hbase = H + (size_t)row0 * 64;
#pragma unroll
  for (int i = 0; i < 8; ++i) {
    const size_t r = (size_t)(i + 8 * half) * 64;
    hbase[r +      m] = acc0[i];
    hbase[r + 16 + m] = acc1[i];
    hbase[r + 32 + m] = acc2[i];
    hbase[r + 48 + m] = acc3[i];
  }
}

// ---- edge scatter: agg[dst] += h[src] * dinv[src]*dinv[dst] ---------------
// 2 features per thread; hardware f32 atomics (L2-resident working set).
__global__ void scatter_kernel(const long long* __restrict__ ei,
                               const float* __restrict__ dinv,
                               const float* __restrict__ H,
                               float* __restrict__ agg, long long E) {
  long long t = (long long)blockIdx.x * blockDim.x + threadIdx.x;
  if (t >= E * 32) return;
  const long long e = t >> 5;
  const int f = (int)(t & 31) * 2;
  const long long s = ei[e];
  const long long d = ei[E + e];
  const float coef = dinv[s] * dinv[d];
  const float2 hv = *reinterpret_cast<const float2*>(H + s * 64 + f);
  float* ap = agg + d * 64 + f;
  unsafeAtomicAdd(ap,     hv.x * coef);
  unsafeAtomicAdd(ap + 1, hv.y * coef);
}

// ---- epilogue: out = agg + h*dinv^2 + b, then (optional) BN + ReLU --------
__global__ void post_kernel(const float4* __restrict__ agg,
                            const float4* __restrict__ h,
                            const float* __restrict__ dinv,
                            const float* __restrict__ bias,
                            const float* __restrict__ g,
                            const float* __restrict__ beta,
                            const float* __restrict__ rm,
                            const float* __restrict__ rv,
                            float4* __restrict__ out,
                            long long nquads, int bn) {
  long long t = (long long)blockIdx.x * blockDim.x + threadIdx.x;
  if (t >= nquads) return;
  const int node = (int)(t >> 4);
  const int f0 = (int)(t & 15) * 4;
  float sc = dinv[node];
  sc *= sc;
  const float4 a = agg[t];
  const float4 hh = h[t];
  float va[4] = {a.x, a.y, a.z, a.w};
  float vh[4] = {hh.x, hh.y, hh.z, hh.w};
  float vo[4];
#pragma unroll
  for (int j = 0; j < 4; ++j) {
    float v = va[j] + vh[j] * sc + bias[f0 + j];
    if (bn) {
      v = (v - rm[f0 + j]) * rsqrtf(rv[f0 + j] + BN_EPS) * g[f0 + j] + beta[f0 + j];
      v = fmaxf(v, 0.0f);
    }
    vo[j] = v;
  }
  out[t] = make_float4(vo[0], vo[1], vo[2], vo[3]);
}

// ---------------------------------------------------------------------------
extern "C" void kernel_launch(void* const* d_in, const int* in_sizes, int n_in,
                              void* d_out, int out_size, void* d_ws, size_t ws_size,
                              hipStream_t stream) {
  (void)n_in; (void)out_size; (void)ws_size;

  const float*     x   = (const float*)d_in[0];
  const long long* ei  = (const long long*)d_in[1];   // [2, E] int64
  const float* W1 = (const float*)d_in[2];
  const float* b1 = (const float*)d_in[3];
  const float* W2 = (const float*)d_in[4];
  const float* b2 = (const float*)d_in[5];
  const float* W3 = (const float*)d_in[6];
  const float* b3 = (const float*)d_in[7];
  const float* g1  = (const float*)d_in[8];
  const float* be1 = (const float*)d_in[9];
  const float* rm1 = (const float*)d_in[10];
  const float* rv1 = (const float*)d_in[11];
  const float* g2  = (const float*)d_in[12];
  const float* be2 = (const float*)d_in[13];
  const float* rm2 = (const float*)d_in[14];
  const float* rv2 = (const float*)d_in[15];

  const int N = in_sizes[0] / 128;               // 100000
  const long long E = (long long)in_sizes[1] / 2;

  // workspace layout (floats): dinv[N] | h[N*64] | agg[N*64] | act[N*64]
  float* dinv   = (float*)d_ws;
  float* bufH   = dinv + N;
  float* bufAgg = bufH + (size_t)N * 64;
  float* bufAct = bufAgg + (size_t)N * 64;
  float* outp   = (float*)d_out;

  const int TB = 256;
  const long long feat4 = (long long)N * 16;     // N*64 floats as float4 quads
  const long long sThreads = E * 32;             // 2 feats per thread
  const unsigned zDegBlocks  = (unsigned)(((long long)N / 4 + TB - 1) / TB);
  const unsigned degBlocks   = (unsigned)((E + TB - 1) / TB);
  const unsigned dinvBlocks  = (unsigned)((N + TB - 1) / TB);
  const unsigned zAggBlocks  = (unsigned)((feat4 + TB - 1) / TB);
  const unsigned scatBlocks  = (unsigned)((sThreads + TB - 1) / TB);
  const unsigned postBlocks  = zAggBlocks;
  const int tiles = (N + 15) / 16;
  const unsigned gemmBlocks  = (unsigned)((tiles + 7) / 8);   // 8 waves/block

  // ---- normalization coefficients ----
  zero_f32_kernel<<<zDegBlocks, TB, 0, stream>>>(dinv, (long long)N / 4);
  deg_kernel<<<degBlocks, TB, 0, stream>>>(ei, dinv, E);
  dinv_kernel<<<dinvBlocks, TB, 0, stream>>>(dinv, N);

  // ---- layer 1: IN=128 -> H=64, BN+ReLU ----
  gcn_gemm_kernel<128><<<gemmBlocks, 256, 0, stream>>>(x, W1, bufH, N);
  zero_f32_kernel<<<zAggBlocks, TB, 0, stream>>>(bufAgg, feat4);
  scatter_kernel<<<scatBlocks, TB, 0, stream>>>(ei, dinv, bufH, bufAgg, E);
  post_kernel<<<postBlocks, TB, 0, stream>>>((const float4*)bufAgg, (const float4*)bufH,
                                             dinv, b1, g1, be1, rm1, rv1,
                                             (float4*)bufAct, feat4, 1);

  // ---- layer 2: H=64 -> H=64, BN+ReLU ----
  gcn_gemm_kernel<64><<<gemmBlocks, 256, 0, stream>>>(bufAct, W2, bufH, N);
  zero_f32_kernel<<<zAggBlocks, TB, 0, stream>>>(bufAgg, feat4);
  scatter_kernel<<<scatBlocks, TB, 0, stream>>>(ei, dinv, bufH, bufAgg, E);
  post_kernel<<<postBlocks, TB, 0, stream>>>((const float4*)bufAgg, (const float4*)bufH,
                                             dinv, b2, g2, be2, rm2, rv2,
                                             (float4*)bufAct, feat4, 1);

  // ---- layer 3: H=64 -> OUT=64, no BN/ReLU ----
  gcn_gemm_kernel<64><<<gemmBlocks, 256, 0, stream>>>(bufAct, W3, bufH, N);
  zero_f32_kernel<<<zAggBlocks, TB, 0, stream>>>(bufAgg, feat4);
  scatter_kernel<<<scatBlocks, TB, 0, stream>>>(ei, dinv, bufH, bufAgg, E);
  post_kernel<<<postBlocks, TB, 0, stream>>>((const float4*)bufAgg, (const float4*)bufH,
                                             dinv, b3, b3, b3, b3, b3,
                                             (float4*)outp, feat4, 0);
}